// GELU205_39857296507243
// MI455X (gfx1250) — compile-verified
//
#include <hip/hip_runtime.h>
#include <hip/hip_bf16.h>
#include <stdint.h>

#define D_DIM    2048
#define NTHREADS 256
#define VPT      8      // D_DIM / NTHREADS
#define KSEL     64

typedef __attribute__((ext_vector_type(4))) float v4f;

__device__ __forceinline__ float fast_tanh(float u) {
    u = fminf(10.0f, fmaxf(-10.0f, u));
    float e = __expf(2.0f * u);
    return (e - 1.0f) / (e + 1.0f);
}

__global__ __launch_bounds__(NTHREADS)
void gelu205_kernel(const float* __restrict__ x,
                    const float* __restrict__ p_log_tau,
                    const float* __restrict__ p_log_beta,
                    const float* __restrict__ p_log_gamma,
                    const float* __restrict__ ema_mean,
                    const float* __restrict__ ema_sq,
                    const float* __restrict__ ema_out_mean,
                    const float* __restrict__ ema_out_sq,
                    const float* __restrict__ ema_out_dir,
                    float* __restrict__ outp)
{
    __shared__ __align__(16) float sx[D_DIM];   // staged input row
    __shared__ unsigned s_hist[512];            // dual 256-bin histograms
    __shared__ unsigned s_prefix[2];
    __shared__ unsigned s_kleft[2];
    __shared__ float    s_red[24];              // 3 reductions x 8 waves
    __shared__ float    s_gcos;

    const int tid = threadIdx.x;
    const int c0  = tid * VPT;
    const long long row = blockIdx.x;
    const float* gx = x + row * (long long)D_DIM;

    // ---- CDNA5 async path: stage the 8KB row into LDS, non-temporal
    //      (x is streamed once; keep L2 for the re-used channel stats) ----
    {
        uint32_t lds0 = (uint32_t)(uintptr_t)(&sx[0]) + (uint32_t)(tid * 16);
        uint32_t lds1 = lds0 + NTHREADS * 16;
        uint64_t g0 = (uint64_t)(uintptr_t)(gx + tid * 4);
        uint64_t g1 = g0 + (uint64_t)(NTHREADS * 16);
        asm volatile("global_load_async_to_lds_b128 %0, %1, off th:TH_LOAD_NT"
                     :: "v"(lds0), "v"(g0) : "memory");
        asm volatile("global_load_async_to_lds_b128 %0, %1, off th:TH_LOAD_NT"
                     :: "v"(lds1), "v"(g1) : "memory");
    }

    // ---- overlap: channel stats (L2-resident, regular-temporal) + scalars ----
    float mean8[VPT], sq8[VPT], omean8[VPT], osq8[VPT], dir8[VPT];
    *(float4*)(mean8)      = *(const float4*)(ema_mean     + c0);
    *(float4*)(mean8 + 4)  = *(const float4*)(ema_mean     + c0 + 4);
    *(float4*)(sq8)        = *(const float4*)(ema_sq       + c0);
    *(float4*)(sq8 + 4)    = *(const float4*)(ema_sq       + c0 + 4);
    *(float4*)(omean8)     = *(const float4*)(ema_out_mean + c0);
    *(float4*)(omean8 + 4) = *(const float4*)(ema_out_mean + c0 + 4);
    *(float4*)(osq8)       = *(const float4*)(ema_out_sq   + c0);
    *(float4*)(osq8 + 4)   = *(const float4*)(ema_out_sq   + c0 + 4);
    *(float4*)(dir8)       = *(const float4*)(ema_out_dir  + c0);
    *(float4*)(dir8 + 4)   = *(const float4*)(ema_out_dir  + c0 + 4);

    const float tau = __expf(p_log_tau[0]);
    const float lb = p_log_beta[0];
    const float beta  = (lb > 20.0f) ? lb : log1pf(__expf(lb));
    const float lg = p_log_gamma[0];
    const float gamma = (lg > 20.0f) ? lg : log1pf(__expf(lg));

    if (tid < 2) { s_prefix[tid] = 0u; s_kleft[tid] = KSEL; }

    // drain this wave's async loads; barrier makes the whole row visible
    asm volatile("s_wait_asynccnt 0x0" ::: "memory");
    __syncthreads();

    // ---- read my 8 channels from LDS, compute gelu + z-scores ----
    float xv[VPT], ov[VPT], zin[VPT], zout[VPT];
    {
        float4 a = *(const float4*)(&sx[c0]);
        float4 b = *(const float4*)(&sx[c0 + 4]);
        xv[0]=a.x; xv[1]=a.y; xv[2]=a.z; xv[3]=a.w;
        xv[4]=b.x; xv[5]=b.y; xv[6]=b.z; xv[7]=b.w;
    }
    const float GC = 0.7978845608028654f;
    float pd = 0.0f, ps = 0.0f, pn = 0.0f;
#pragma unroll
    for (int j = 0; j < VPT; ++j) {
        float xj = xv[j];
        float o  = 0.5f * xj * (1.0f + fast_tanh(GC * (xj + 0.044715f * xj * xj * xj)));
        ov[j] = o;
        float vi = fmaxf(sq8[j]  - mean8[j]  * mean8[j],  1e-4f);
        zin[j]  = (xj - mean8[j])  / (sqrtf(vi) + 1e-5f);
        float vo = fmaxf(osq8[j] - omean8[j] * omean8[j], 1e-4f);
        zout[j] = (o  - omean8[j]) / (sqrtf(vo) + 1e-5f);
        pd += o * dir8[j];
        ps += o * o;
        pn += dir8[j] * dir8[j];
    }

    // ---- wave32 reductions -> LDS -> cosine gate ----
    const int lane = tid & 31, wv = tid >> 5;
#pragma unroll
    for (int off = 16; off > 0; off >>= 1) {
        pd += __shfl_down(pd, off);
        ps += __shfl_down(ps, off);
        pn += __shfl_down(pn, off);
    }
    if (lane == 0) { s_red[wv] = pd; s_red[8 + wv] = ps; s_red[16 + wv] = pn; }
    __syncthreads();
    if (tid == 0) {
        float dot = 0.0f, ss = 0.0f, nd = 0.0f;
#pragma unroll
        for (int w = 0; w < 8; ++w) { dot += s_red[w]; ss += s_red[8 + w]; nd += s_red[16 + w]; }
        float no  = fmaxf(sqrtf(ss), 1e-12f);
        float ndn = fmaxf(sqrtf(nd), 1e-12f);
        float cs  = fminf(1.0f, fmaxf(-1.0f, dot / (no * ndn)));
        s_gcos = __expf(-tau * cs);
    }

    // ---- dual radix select: exact 64th-largest of |zin| and |zout| ----
    for (int p = 3; p >= 0; --p) {
        const int shift = 8 * p;
        s_hist[tid] = 0u; s_hist[tid + 256] = 0u;
        __syncthreads();
        const unsigned pin = s_prefix[0], pout = s_prefix[1];
#pragma unroll
        for (int j = 0; j < VPT; ++j) {
            unsigned ki = __float_as_uint(fabsf(zin[j]));
            unsigned ko = __float_as_uint(fabsf(zout[j]));
            if (p == 3 || ((ki ^ pin)  >> (shift + 8)) == 0u)
                atomicAdd(&s_hist[(ki >> shift) & 0xFFu], 1u);
            if (p == 3 || ((ko ^ pout) >> (shift + 8)) == 0u)
                atomicAdd(&s_hist[256 + ((ko >> shift) & 0xFFu)], 1u);
        }
        __syncthreads();
        // cooperative wave32 scan: wave0 -> |zin| hist, wave1 -> |zout| hist
        if (wv < 2) {
            const int sel = wv;
            unsigned c[8], lsum = 0u;
            const int base = sel * 256 + lane * 8;
#pragma unroll
            for (int i = 0; i < 8; ++i) { c[i] = s_hist[base + i]; lsum += c[i]; }
            // inclusive suffix-sum across lanes (acc[L] = sum of lanes >= L)
            unsigned acc = lsum;
#pragma unroll
            for (int off = 1; off < 32; off <<= 1) {
                unsigned o = __shfl_down(acc, off);
                if (lane + off < 32) acc += o;
            }
            const unsigned above = acc - lsum;      // keys in bins of higher lanes
            const unsigned kleft = s_kleft[sel];
            unsigned cum = above;
#pragma unroll
            for (int i = 7; i >= 0; --i) {          // descend within my 8 bins
                unsigned cc = c[i];
                if (cum < kleft && cum + cc >= kleft) {
                    s_prefix[sel] |= (unsigned)(lane * 8 + i) << shift;
                    s_kleft[sel] = kleft - cum;
                }
                cum += cc;
            }
        }
        __syncthreads();
    }

    const unsigned thr_in = s_prefix[0], thr_out = s_prefix[1];
    const float gcos = s_gcos;

    // ---- gate + non-temporal streaming store ----
    float res[VPT];
#pragma unroll
    for (int j = 0; j < VPT; ++j) {
        unsigned ki = __float_as_uint(fabsf(zin[j]));
        unsigned ko = __float_as_uint(fabsf(zout[j]));
        float g = 1.0f;
        if (ki >= thr_in && ko >= thr_out)
            g = fminf(8.0f, fmaxf(0.1f, 1.0f + beta * fast_tanh(gamma * zin[j])));
        res[j] = ov[j] * g * gcos;
    }
    v4f r0 = { res[0], res[1], res[2], res[3] };
    v4f r1 = { res[4], res[5], res[6], res[7] };
    v4f* orow = (v4f*)(outp + row * (long long)D_DIM + c0);
    __builtin_nontemporal_store(r0, orow);
    __builtin_nontemporal_store(r1, orow + 1);
}

extern "C" void kernel_launch(void* const* d_in, const int* in_sizes, int n_in,
                              void* d_out, int out_size, void* d_ws, size_t ws_size,
                              hipStream_t stream) {
    const float* x            = (const float*)d_in[0];
    // d_in[1] = logit_decay: affects only EMA side-state, not the output.
    const float* log_tau      = (const float*)d_in[2];
    const float* log_beta     = (const float*)d_in[3];
    const float* log_gamma    = (const float*)d_in[4];
    const float* ema_mean     = (const float*)d_in[5];
    const float* ema_sq       = (const float*)d_in[6];
    const float* ema_out_mean = (const float*)d_in[7];
    const float* ema_out_sq   = (const float*)d_in[8];
    const float* ema_out_dir  = (const float*)d_in[9];
    float* outp = (float*)d_out;

    const int rows = in_sizes[0] / D_DIM;   // B*T = 16384
    gelu205_kernel<<<rows, NTHREADS, 0, stream>>>(
        x, log_tau, log_beta, log_gamma,
        ema_mean, ema_sq, ema_out_mean, ema_out_sq, ema_out_dir, outp);
}